// CondMultiChannel2DCircularConv_70385924047443
// MI455X (gfx1250) — compile-verified
//
#include <hip/hip_runtime.h>
#include <hip/hip_bf16.h>

// ---------------------------------------------------------------------------
// CondMultiChannel2DCircularConv for MI455X (gfx1250, wave32, WMMA)
//   B=8, C=64, N=32, K=3
//   d_in[0] conv_in   [8,64,32,32] f32
//   d_in[1] pre_kernel[8,64,64,3,3] f32
//   d_in[2] bias      [8,64,1,1]   f32
//   d_out: conv_out (524288 f32) ++ logdet (8 f32)
//   d_ws : Kmat transposed [8][64][9][64] f32 = 1,179,648 bytes
// ---------------------------------------------------------------------------

typedef __attribute__((ext_vector_type(2))) float v2f;
typedef __attribute__((ext_vector_type(8))) float v8f;

#define BB 8
#define CC 64
#define NN 32
#define CONV_OUT_ELEMS (BB*CC*NN*NN)   // 524288

// ---------------------------------------------------------------------------
// Stage 1: kt[b][o][tap][i] = 0.7*tanh(pre[b][o][i][tap]) + (o==i && tap==4)
// Tap-major reduction layout keeps WMMA A-fragment loads contiguous and
// makes `tap` a uniform loop constant in the conv kernel (4 | 64).
// ---------------------------------------------------------------------------
__global__ void build_kmat_kernel(const float* __restrict__ pre,
                                  float* __restrict__ kt, int total) {
    int idx = blockIdx.x * blockDim.x + threadIdx.x;   // [b][o][i][tap] order
    if (idx >= total) return;
    float v = 0.7f * tanhf(pre[idx]);
    int tap = idx % 9;
    int rem = idx / 9;
    int i  = rem & 63;
    int bo = rem >> 6;                  // b*64 + o
    int o  = bo & 63;
    if (o == i && tap == 4) v += 1.0f;  // identity kernel center tap
    kt[((size_t)bo * 9 + tap) * CC + i] = v;
}

// ---------------------------------------------------------------------------
// Stage 2: circular conv as GEMM on V_WMMA_F32_16X16X4_F32.
//   Block = 128 threads (4 waves), one block per (b, u, v0) pixel tile.
//   Wave w computes o-rows w*16..w*16+15; all 4 waves share the LDS-staged
//   circular halo slab x[b, :, u-1..u+1, v0-1..v0+16]  (64 x 3 x 18 f32),
//   staged via GLOBAL_LOAD_ASYNC_TO_LDS_B32 (ASYNCcnt-tracked DMA).
//   Reduction k = tap*64 + ci:  D[o,n] += A[o,k] * B[k,n],
//     A[o,k] = kt[b][o][tap][ci],  B[k,n] = xs[ci][2-dy][n+2-dx].
// Fragment layouts per ISA 7.12.2 (32-bit A 16x4, 32-bit C/D 16x16).
// ---------------------------------------------------------------------------
#define HALO (CC * 3 * 18)              // 3456 floats = 27*128
__global__ __launch_bounds__(128) void conv_wmma_kernel(
    const float* __restrict__ x,        // [8,64,32,32]
    const float* __restrict__ kt,       // [8,64,9,64]
    const float* __restrict__ bias,     // [8,64]
    float* __restrict__ out)            // [8,64,32,32]
{
    __shared__ float xs[HALO];          // [ci][hr][hc]  13824 B

    const int tid  = threadIdx.x;
    const int wave = tid >> 5;          // o-tile 0..3
    const int lane = tid & 31;
    const int ln   = lane & 15;         // A-frag M / D-frag N index
    const int hi   = lane >> 4;         // lane-half selector

    int blk = blockIdx.x;               // 0..511
    int b   = blk >> 6;
    int pt  = blk & 63;
    int u   = pt >> 1;                  // output row
    int v0  = (pt & 1) << 4;            // output col base

    // ---- async-DMA the circular halo slab into LDS (27 b32/lane) ----------
    // Per-lane global gather address + per-lane LDS dest; tracked by ASYNCcnt
    // (flat->LDS offset = low 32 bits of the generic pointer, ISA 10.2).
    const float* xb = x + (size_t)b * CC * NN * NN;
    for (int idx = tid; idx < HALO; idx += 128) {
        int ci  = idx / 54;
        int rem = idx - ci * 54;
        int hr  = rem / 18;
        int hc  = rem - hr * 18;
        int row = (u - 1 + hr) & (NN - 1);
        int col = (v0 - 1 + hc) & (NN - 1);
        unsigned     lds_off = (unsigned)(size_t)&xs[idx];
        const float* gp      = xb + (ci * NN + row) * NN + col;
        asm volatile("global_load_async_to_lds_b32 %0, %1, off"
                     :: "v"(lds_off), "v"(gp) : "memory");
    }
    asm volatile("s_wait_asynccnt 0" ::: "memory");   // this wave's DMAs done
    __syncthreads();                                   // publish across waves

    // ---- 144 WMMA k-steps: 9 taps (unrolled, const offsets) x 16 chunks ----
    const float* Arow = kt + ((size_t)(b * CC) + wave * 16 + ln) * 9 * CC;
    __builtin_prefetch(Arow, 0, 0);     // global_prefetch_b8

    v8f acc = {};
#pragma unroll
    for (int tap = 0; tap < 9; ++tap) {
        const int dy = tap / 3, dx = tap - dy * 3;          // compile-time
        const float* At = Arow + tap * CC;
        const int bbase = (2 - dy) * 18 + (ln + 2 - dx);    // halo offset
        for (int kc = 0; kc < 16; ++kc) {
            int ci0 = kc * 4 + hi * 2;  // VGPR j holds ci = ci0 + j
            v2f a, bf;
            a[0]  = At[ci0];
            a[1]  = At[ci0 + 1];
            bf[0] = xs[ci0 * 54 + bbase];
            bf[1] = xs[(ci0 + 1) * 54 + bbase];
            acc = __builtin_amdgcn_wmma_f32_16x16x4_f32(
                /*neg_a=*/false, a, /*neg_b=*/false, bf,
                /*c_mod=*/(short)0, acc, /*reuse_a=*/false, /*reuse_b=*/false);
        }
    }

    // ---- store D (+bias): VGPR r -> row M = r + 8*hi, col N = ln ----
    float* ob = out + (size_t)((b * CC) + wave * 16) * (NN * NN) + u * NN + v0 + ln;
    const float* bs = bias + b * CC + wave * 16;
#pragma unroll
    for (int r = 0; r < 8; ++r) {
        int mo = r + 8 * hi;
        ob[(size_t)mo * (NN * NN)] = acc[r] + bs[mo];
    }
}

// ---------------------------------------------------------------------------
// Stage 3a: zero the 8 logdet accumulators (harness poisons d_out).
// ---------------------------------------------------------------------------
__global__ void init_logdet_kernel(float* __restrict__ ld) {
    int i = threadIdx.x;
    if (i < BB) ld[i] = 0.0f;
}

// ---------------------------------------------------------------------------
// Stage 3b: log|det K_hat(u,v)| per (b,u,v), summed into logdet[b].
//   K_hat[o][i] = sum_tap kt[b][o][tap][i] * exp(-2*pi*j*(u*(dy-1)+v*(dx-1))/32)
//   LU with partial pivoting in LDS; row stride 65 -> conflict-free on the
//   64-bank LDS; elimination uses 4 threads per row (252 active lanes).
// ---------------------------------------------------------------------------
#define LDA 65
__global__ __launch_bounds__(256) void logdet_kernel(
    const float* __restrict__ kt,       // [8,64,9,64]
    float* __restrict__ logdet)         // [8]
{
    __shared__ float Are[CC * LDA];
    __shared__ float Aim[CC * LDA];
    __shared__ int   s_piv;

    int bid = blockIdx.x;               // 0..8191
    int b   = bid >> 10;
    int uv  = bid & 1023;
    int u   = uv >> 5;
    int v   = uv & 31;
    int t   = threadIdx.x;

    // 9 twiddles (uniform across the workgroup)
    float wr[9], wi[9];
#pragma unroll
    for (int tap = 0; tap < 9; ++tap) {
        int dy = tap / 3, dx = tap - dy * 3;
        float ph = -2.0f * 3.14159265358979323846f *
                   (float)(u * (dy - 1) + v * (dx - 1)) / 32.0f;
        float s, c;
        sincosf(ph, &s, &c);
        wr[tap] = c;
        wi[tap] = s;
    }

    // Build K_hat into LDS (consecutive threads -> consecutive i: coalesced)
    const float* kb = kt + (size_t)b * CC * 9 * CC;
    for (int e = t; e < CC * CC; e += 256) {
        int o = e >> 6, i = e & 63;
        const float* kp = kb + (size_t)o * 9 * CC + i;
        float sre = 0.0f, sim = 0.0f;
#pragma unroll
        for (int tap = 0; tap < 9; ++tap) {
            float kv = kp[tap * CC];
            sre = fmaf(kv, wr[tap], sre);
            sim = fmaf(kv, wi[tap], sim);
        }
        Are[o * LDA + i] = sre;
        Aim[o * LDA + i] = sim;
    }
    __syncthreads();

    float logsum = 0.0f;                // meaningful on thread 0 only
    const int tr = t >> 2;              // row offset (4 threads per row)
    const int tc = t & 3;               // column phase

    for (int k = 0; k < CC; ++k) {
        // partial pivot: max |re|+|im| in column k, rows k..63
        if (t == 0) {
            int p = k;
            float best = -1.0f;
            for (int r = k; r < CC; ++r) {
                float m = fabsf(Are[r * LDA + k]) + fabsf(Aim[r * LDA + k]);
                if (m > best) { best = m; p = r; }
            }
            s_piv = p;
        }
        __syncthreads();
        int p = s_piv;
        if (p != k) {                   // uniform branch
            for (int c = k + t; c < CC; c += 256) {
                float a0 = Are[k * LDA + c]; Are[k * LDA + c] = Are[p * LDA + c]; Are[p * LDA + c] = a0;
                float i0 = Aim[k * LDA + c]; Aim[k * LDA + c] = Aim[p * LDA + c]; Aim[p * LDA + c] = i0;
            }
        }
        __syncthreads();

        float pr = Are[k * LDA + k], pi = Aim[k * LDA + k];
        float mag2 = fmaxf(pr * pr + pi * pi, 1e-37f);
        if (t == 0) logsum += 0.5f * logf(mag2);

        float invd = 1.0f / mag2;
        float ir = pr * invd, ii = -pi * invd;     // 1/pivot

        int r = k + 1 + tr;             // 4 threads cooperate per row
        if (r < CC) {
            float ar = Are[r * LDA + k], ai = Aim[r * LDA + k];
            float mr = ar * ir - ai * ii;
            float mi = ar * ii + ai * ir;
            for (int c = k + 1 + tc; c < CC; c += 4) {
                float kr = Are[k * LDA + c], ki = Aim[k * LDA + c];
                Are[r * LDA + c] -= mr * kr - mi * ki;
                Aim[r * LDA + c] -= mr * ki + mi * kr;
            }
        }
        __syncthreads();
    }

    if (t == 0) atomicAdd(&logdet[b], logsum);     // global_atomic_add_f32
}

// ---------------------------------------------------------------------------
extern "C" void kernel_launch(void* const* d_in, const int* in_sizes, int n_in,
                              void* d_out, int out_size, void* d_ws, size_t ws_size,
                              hipStream_t stream) {
    const float* conv_in = (const float*)d_in[0];   // [8,64,32,32]
    const float* pre_k   = (const float*)d_in[1];   // [8,64,64,3,3]
    const float* bias    = (const float*)d_in[2];   // [8,64,1,1]
    float* out    = (float*)d_out;                  // conv_out ++ logdet
    float* logdet = out + CONV_OUT_ELEMS;
    float* kt     = (float*)d_ws;                   // [8][64][9][64] = 1.18 MB

    // Stage 1: materialize modified kernel (tap-major, shared by conv+logdet)
    int kmat_total = BB * CC * CC * 9;              // 294912
    build_kmat_kernel<<<(kmat_total + 255) / 256, 256, 0, stream>>>(pre_k, kt, kmat_total);

    // Stage 3a: zero logdet accumulators
    init_logdet_kernel<<<1, 32, 0, stream>>>(logdet);

    // Stage 2: WMMA circular conv (512 blocks x 4 waves, async-LDS B slab)
    conv_wmma_kernel<<<BB * 64, 128, 0, stream>>>(conv_in, kt, bias, out);

    // Stage 3b: per-frequency complex LU -> logdet (8192 workgroups)
    logdet_kernel<<<BB * NN * NN, 256, 0, stream>>>(kt, logdet);
}